// SPDCov2D_16595753632502
// MI455X (gfx1250) — compile-verified
//
#include <hip/hip_runtime.h>
#include <math.h>

// Fréchet (Karcher) mean scan over SPD matrices, matmul-only formulation.
// B=512 batches, N=16 matrices per batch, D=64. One workgroup per batch.
// All matrix functions (sqrt, invsqrt, log, exp, fractional power) are
// computed with Newton-Schulz / series iterations whose only heavy op is a
// 64x64 fp32 matmul executed on V_WMMA_F32_16X16X4_F32.

#define DD    64
#define NMAT  16
#define LDSS  66          // padded LDS row stride (floats)
#define TPB   128         // 4 waves of 32

typedef __attribute__((ext_vector_type(2))) float v2f;
typedef __attribute__((ext_vector_type(8))) float v8f;

// ---------------------------------------------------------------------------
// 64x64 fp32 matmul on WMMA: C = A * B.  A,B,C are LDS matrices with row
// stride LDSS.  Wave w owns output row-block [16w,16w+16); it keeps 4
// accumulators (one per 16-col block) live and walks K outermost so the A
// fragment is loaded ONCE per k-step and fed to 4 chained WMMAs.
//
// Fragment layouts per CDNA5 ISA 7.12.2:
//   A (16x4 fp32): lanes 0-15 rows M=0..15, a[0]=K0,a[1]=K1; lanes 16-31 K2,K3
//   B (4x16 fp32): lanes 0-15 cols N=0..15, b[0]=row K0,b[1]=row K1; lanes 16-31 rows K2,K3
//   C/D (16x16 fp32): VGPR r -> (M=r, N=lane) lanes 0-15 ; (M=r+8, N=lane-16)
// ---------------------------------------------------------------------------
__device__ __forceinline__ void mm64(float* __restrict__ C,
                                     const float* __restrict__ A,
                                     const float* __restrict__ B,
                                     int lane, int wave)
{
    const int row  = lane & 15;
    const int half = lane >> 4;
    const int ti   = wave << 4;                    // this wave's output row block

    v8f acc0 = {0.f,0.f,0.f,0.f,0.f,0.f,0.f,0.f};
    v8f acc1 = acc0, acc2 = acc0, acc3 = acc0;

    #pragma unroll
    for (int k4 = 0; k4 < DD; k4 += 4) {
        // one A fragment per k-step, shared by all 4 column-block WMMAs
        const float* ap = A + (ti + row) * LDSS + (k4 + 2 * half);
        v2f a;
        a[0] = ap[0];                              // K = k4 + 2*half
        a[1] = ap[1];                              // K = k4 + 2*half + 1

        const float* bp = B + (k4 + 2 * half) * LDSS + row;
        v2f b0, b1, b2, b3;
        b0[0] = bp[0];        b0[1] = bp[LDSS];        // col block 0
        b1[0] = bp[16];       b1[1] = bp[LDSS + 16];   // col block 1
        b2[0] = bp[32];       b2[1] = bp[LDSS + 32];   // col block 2
        b3[0] = bp[48];       b3[1] = bp[LDSS + 48];   // col block 3

        acc0 = __builtin_amdgcn_wmma_f32_16x16x4_f32(false, a, false, b0,
                                                     (short)0, acc0, false, false);
        acc1 = __builtin_amdgcn_wmma_f32_16x16x4_f32(false, a, false, b1,
                                                     (short)0, acc1, false, false);
        acc2 = __builtin_amdgcn_wmma_f32_16x16x4_f32(false, a, false, b2,
                                                     (short)0, acc2, false, false);
        acc3 = __builtin_amdgcn_wmma_f32_16x16x4_f32(false, a, false, b3,
                                                     (short)0, acc3, false, false);
    }

    const int cm = ti + half * 8;
    #pragma unroll
    for (int r = 0; r < 8; ++r) {
        C[(cm + r) * LDSS + (row +  0)] = acc0[r];
        C[(cm + r) * LDSS + (row + 16)] = acc1[r];
        C[(cm + r) * LDSS + (row + 32)] = acc2[r];
        C[(cm + r) * LDSS + (row + 48)] = acc3[r];
    }
    __syncthreads();
}

// ---------------- element-wise helpers (each ends in a barrier) -------------
__device__ __forceinline__ void ew_scaleI(float* __restrict__ Dm,
                                          const float* __restrict__ Sm,
                                          float a, float dg, int tid)
{   // D = a*S + dg*I
    for (int i = tid; i < DD * DD; i += TPB) {
        const int r = i >> 6, c = i & 63;
        Dm[r * LDSS + c] = a * Sm[r * LDSS + c] + ((r == c) ? dg : 0.0f);
    }
    __syncthreads();
}

__device__ __forceinline__ void set_identity(float* __restrict__ Dm, int tid)
{
    for (int i = tid; i < DD * DD; i += TPB) {
        const int r = i >> 6, c = i & 63;
        Dm[r * LDSS + c] = (r == c) ? 1.0f : 0.0f;
    }
    __syncthreads();
}

__device__ __forceinline__ void ew_newton(float* __restrict__ Tm, int tid)
{   // T = (3I - T)/2   (Newton-Schulz polynomial, in place)
    for (int i = tid; i < DD * DD; i += TPB) {
        const int r = i >> 6, c = i & 63;
        const float v = Tm[r * LDSS + c];
        Tm[r * LDSS + c] = 0.5f * (((r == c) ? 3.0f : 0.0f) - v);
    }
    __syncthreads();
}

__device__ __forceinline__ void ew_symscale(float* __restrict__ Dm,
                                            const float* __restrict__ Sm,
                                            float a, int tid)
{   // D = a*(S + S^T)   (D must differ from S)
    for (int i = tid; i < DD * DD; i += TPB) {
        const int r = i >> 6, c = i & 63;
        Dm[r * LDSS + c] = a * (Sm[r * LDSS + c] + Sm[c * LDSS + r]);
    }
    __syncthreads();
}

__device__ __forceinline__ float trace64(const float* __restrict__ Am,
                                         float* __restrict__ red, int tid)
{
    if (tid < DD) red[tid] = Am[tid * LDSS + tid];
    __syncthreads();
    if (tid == 0) {
        float s = 0.0f;
        #pragma unroll
        for (int i = 0; i < DD; ++i) s += red[i];
        red[0] = s;
    }
    __syncthreads();
    const float s = red[0];
    __syncthreads();
    return s;
}

// ---------------------------------------------------------------------------
// Coupled Newton-Schulz: given SPD A with spectrum of A*invc in (0,1],
// iterate Y<-Y(3I-ZY)/2, Z<-(3I-ZY)Z/2  =>  Y -> (A*invc)^{1/2},
// Z -> (A*invc)^{-1/2}.  Ping-pongs between (y0,z0) and (y1,z1); T is scratch.
// ---------------------------------------------------------------------------
__device__ __forceinline__ void ns_sqrt(const float* __restrict__ Ain, float invc,
                                        int iters,
                                        float* y0, float* y1,
                                        float* z0, float* z1, float* T,
                                        float** Yout, float** Zout,
                                        int tid, int lane, int wave)
{
    ew_scaleI(y0, Ain, invc, 0.0f, tid);
    set_identity(z0, tid);
    float *Yc = y0, *Yn = y1, *Zc = z0, *Zn = z1;
    for (int it = 0; it < iters; ++it) {
        mm64(T, Zc, Yc, lane, wave);   // T  = Z*Y
        ew_newton(T, tid);             // T  = (3I - T)/2
        mm64(Yn, Yc, T, lane, wave);   // Y' = Y*T
        mm64(Zn, T, Zc, lane, wave);   // Z' = T*Z
        float* s;
        s = Yc; Yc = Yn; Yn = s;
        s = Zc; Zc = Zn; Zn = s;
    }
    *Yout = Yc;
    *Zout = Zc;
}

// ---------------------------------------------------------------------------
__global__ __launch_bounds__(TPB)
void frechet_mean_kernel(const float* __restrict__ x,
                         const float* __restrict__ wm,
                         float* __restrict__ out)
{
    __shared__ float BM[DD * LDSS], BX[DD * LDSS], BY[DD * LDSS], BZ[DD * LDSS];
    __shared__ float BU[DD * LDSS], BV[DD * LDSS], BT[DD * LDSS], BS[DD * LDSS];
    __shared__ float red[DD];

    const int tid  = (int)threadIdx.x;
    const int lane = tid & 31;
    const int wave = tid >> 5;
    const int b    = (int)blockIdx.x;
    const size_t base = (size_t)b * NMAT * DD * DD;

    // M = x[b, 0]
    for (int i = tid; i < DD * DD; i += TPB)
        BM[(i >> 6) * LDSS + (i & 63)] = x[base + i];
    __syncthreads();

    float w0  = wm[0];
    float cum = w0 * w0;               // un-normalized weights: t_n is scale-invariant

    for (int n = 1; n < NMAT; ++n) {
        float wn = wm[n]; wn = wn * wn;
        cum += wn;
        const float tn = wn / cum;

        // X = x[b, n]
        for (int i = tid; i < DD * DD; i += TPB)
            BX[(i >> 6) * LDSS + (i & 63)] = x[base + (size_t)n * DD * DD + i];
        __syncthreads();

        // --- phase 1: Y=(M/c)^{1/2}, Z=(M/c)^{-1/2};  Ms=sqrt(c)Y, Mis=Z/sqrt(c)
        const float c    = trace64(BM, red, tid);      // spectral radius <= trace
        const float invc = 1.0f / c;
        float *Y, *Z;
        ns_sqrt(BM, invc, 8, BY, BU, BZ, BV, BT, &Y, &Z, tid, lane, wave);

        // --- phase 2: S = Mis*X*Mis = (1/c) * sym(Z X Z)
        mm64(BT, Z, BX, lane, wave);                   // Z*X
        mm64(BV, BT, Z, lane, wave);                   // Z*X*Z
        ew_symscale(BS, BV, 0.5f * invc, tid);         // S

        // --- phase 3: R = (S/c2)^{1/4} via two Newton-Schulz square roots
        const float c2    = trace64(BS, red, tid);
        const float invc2 = 1.0f / c2;
        float *R1, *Zd1, *R, *Zd2;
        ns_sqrt(BS, invc2, 6, BM, BU, BZ, BV, BT, &R1, &Zd1, tid, lane, wave); // R1 in BM
        ns_sqrt(R1, 1.0f, 6, BX, BU, BZ, BV, BT, &R,  &Zd2, tid, lane, wave);  // R  in BX

        // --- phase 4: G = (4*t/8) * log(R), 4-term series in E1 = R - I
        ew_scaleI(BM, R, 1.0f, -1.0f, tid);            // E1
        mm64(BU, BM, BM, lane, wave);                  // E1^2
        mm64(BZ, BU, BM, lane, wave);                  // E1^3
        mm64(BV, BZ, BM, lane, wave);                  // E1^4
        const float gs = tn * 4.0f * 0.125f;           // *4: two sqrts; /8: exp scaling 2^3
        for (int i = tid; i < DD * DD; i += TPB) {
            const int r = i >> 6, cc = i & 63;
            const int o = r * LDSS + cc;
            BT[o] = gs * (BM[o] - 0.5f * BU[o]
                          + (1.0f / 3.0f) * BZ[o] - 0.25f * BV[o]);
        }
        __syncthreads();

        // --- phase 5: E = exp(G): 5th-order Taylor then 3 squarings (G in BT)
        mm64(BM, BT, BT, lane, wave);                  // G^2
        mm64(BU, BM, BT, lane, wave);                  // G^3
        mm64(BZ, BU, BT, lane, wave);                  // G^4
        mm64(BV, BZ, BT, lane, wave);                  // G^5
        for (int i = tid; i < DD * DD; i += TPB) {
            const int r = i >> 6, cc = i & 63;
            const int o = r * LDSS + cc;
            BX[o] = ((r == cc) ? 1.0f : 0.0f) + BT[o] + 0.5f * BM[o]
                    + (1.0f / 6.0f) * BU[o] + (1.0f / 24.0f) * BZ[o]
                    + (1.0f / 120.0f) * BV[o];
        }
        __syncthreads();
        mm64(BM, BX, BX, lane, wave);
        mm64(BX, BM, BM, lane, wave);
        mm64(BM, BX, BX, lane, wave);                  // E in BM

        // --- phase 6: M_new = Ms * P * Ms = (c * c2^t) * sym(Y E Y)
        const float sc = c * powf(c2, tn);
        mm64(BT, BY, BM, lane, wave);                  // Y*E
        mm64(BU, BT, BY, lane, wave);                  // Y*E*Y
        ew_symscale(BM, BU, 0.5f * sc, tid);           // new mean in BM
    }

    for (int i = tid; i < DD * DD; i += TPB)
        out[(size_t)b * DD * DD + i] = BM[(i >> 6) * LDSS + (i & 63)];
}

// ---------------------------------------------------------------------------
extern "C" void kernel_launch(void* const* d_in, const int* in_sizes, int n_in,
                              void* d_out, int out_size, void* d_ws, size_t ws_size,
                              hipStream_t stream)
{
    (void)n_in; (void)out_size; (void)d_ws; (void)ws_size;
    const float* x  = (const float*)d_in[0];          // (B, 16, 64, 64) f32
    const float* wm = (const float*)d_in[1];          // (16,) f32
    float* out      = (float*)d_out;                  // (B, 1, 64, 64) f32
    const int B = in_sizes[0] / (NMAT * DD * DD);
    frechet_mean_kernel<<<dim3(B), dim3(TPB), 0, stream>>>(x, wm, out);
}